// NGMConvLayer_76819785056583
// MI455X (gfx1250) — compile-verified
//
#include <hip/hip_runtime.h>
#include <hip/hip_bf16.h>
#include <math.h>

// ---------------------------------------------------------------------------
// Problem constants (from reference): N=65536 nodes, DIN=DOUT=64, E=4194304,
// sinkhorn over 256x256 with tau=0.05, 10 iterations, n1==n2 (no padding).
// ---------------------------------------------------------------------------
#define NNODES 65536
#define DIM    64
#define SKN    256
#define SK_INV_TAU 20.0f
#define SK_ITERS 10
#define WSTRIDE 66   // padded LDS stride (floats): keeps v2f 8B-aligned, spreads banks

typedef __attribute__((ext_vector_type(2))) float v2f;
typedef __attribute__((ext_vector_type(8))) float v8f;

// --------------------------- degree / dinv ---------------------------------
__global__ void init_deg_kernel(float* __restrict__ deg) {
    int i = blockIdx.x * blockDim.x + threadIdx.x;
    deg[i] = 1.0f;   // self-loop contribution
}

__global__ void deg_count_kernel(const int* __restrict__ eidx,
                                 float* __restrict__ deg, int E) {
    int e = blockIdx.x * blockDim.x + threadIdx.x;
    if (e < E) atomicAdd(&deg[eidx[E + e]], 1.0f);   // dst row of edge_index
}

__global__ void dinv_kernel(float* __restrict__ deg) {
    int i = blockIdx.x * blockDim.x + threadIdx.x;
    float d = deg[i];
    deg[i] = (d > 0.0f) ? rsqrtf(d) : 0.0f;          // in place: deg -> dinv
}

// --------------------------- fused dual WMMA GEMM --------------------------
// One wave32 per 16-row strip of x; computes 16x64 outputs for BOTH weight
// matrices with V_WMMA_F32_16X16X4_F32. Weights staged transposed in LDS so
// each B fragment is one ds_load_b64; A fragments hoisted into registers.
//   xw  <- x @ W_gcn                         (saved for the edge scatter)
//   out <- x @ W_self + b_self + b_gcn + xw * dinv^2   (self-loop folded in)
__global__ void gemm_wmma_kernel(const float* __restrict__ x,
                                 const float* __restrict__ Wg,
                                 const float* __restrict__ bg,
                                 const float* __restrict__ Ws,
                                 const float* __restrict__ bs,
                                 const float* __restrict__ dinv,
                                 float* __restrict__ xw,
                                 float* __restrict__ out) {
    __shared__ float WgT[DIM * WSTRIDE];   // WgT[col][k] = Wg[k][col]
    __shared__ float WsT[DIM * WSTRIDE];

    // Stage both weight matrices transposed (coalesced global reads; every
    // thread runs the same trip count -> no divergence, EXEC stays all-1s).
    for (int idx = threadIdx.x; idx < DIM * DIM; idx += 256) {
        const int k = idx >> 6;        // row of W
        const int c = idx & (DIM - 1); // col of W
        WgT[c * WSTRIDE + k] = Wg[idx];
        WsT[c * WSTRIDE + k] = Ws[idx];
    }
    __syncthreads();

    const int wave = (blockIdx.x * blockDim.x + threadIdx.x) >> 5;
    const int lane = threadIdx.x & 31;
    const int half = lane >> 4;        // 0: K pair {0,1}; 1: K pair {2,3}
    const int lm   = lane & 15;
    const int m0   = wave * 16;        // strip base row (grid sized exactly)

    // Hoist all A fragments: one clause of 16 global b64 loads per lane.
    const int arow = m0 + lm;
    const float* xrow = x + arow * DIM;
    v2f afrag[16];
#pragma unroll
    for (int kb = 0; kb < 16; ++kb)
        afrag[kb] = *(const v2f*)(xrow + kb * 4 + half * 2);

    v8f accg[4] = {};                  // x @ W_gcn, 4 N-tiles of 16
    v8f accs[4] = {};                  // x @ W_self

#pragma unroll
    for (int kb = 0; kb < 16; ++kb) {  // K = 64 in steps of 4
        const int k = kb * 4 + half * 2;
#pragma unroll
        for (int t = 0; t < 4; ++t) {
            const int col = t * 16 + lm;
            v2f bgv = *(const v2f*)&WgT[col * WSTRIDE + k];  // ds_load_b64
            v2f bsv = *(const v2f*)&WsT[col * WSTRIDE + k];
            accg[t] = __builtin_amdgcn_wmma_f32_16x16x4_f32(
                false, afrag[kb], false, bgv, (short)0, accg[t], false, false);
            accs[t] = __builtin_amdgcn_wmma_f32_16x16x4_f32(
                false, afrag[kb], false, bsv, (short)0, accs[t], false, false);
        }
    }

    // C/D layout: VGPR v, lanes 0-15 -> M=v, lanes 16-31 -> M=v+8; N=lane%16.
#pragma unroll
    for (int t = 0; t < 4; ++t) {
        const int col = t * 16 + lm;
#pragma unroll
        for (int v = 0; v < 8; ++v) {
            const int row = m0 + v + half * 8;
            const float g  = accg[t][v];
            const float di = dinv[row];
            xw[row * DIM + col]  = g;
            out[row * DIM + col] = accs[t][v] + bs[col] + bg[col] + g * di * di;
        }
    }
}

// --------------------------- edge scatter ----------------------------------
// One wave per edge; each lane moves 2 floats (b64 gather + 2 f32 atomics).
// xw (16MB) and out (16MB) are L2-resident on MI455X (192MB L2), so the
// atomic RMWs run at L2 rates, not HBM.
__global__ void scatter_kernel(const int* __restrict__ eidx,
                               const float* __restrict__ xw,
                               const float* __restrict__ dinv,
                               float* __restrict__ out, int E) {
    const int gid  = blockIdx.x * blockDim.x + threadIdx.x;
    const int e    = gid >> 5;
    const int lane = gid & 31;
    if (e >= E) return;
    const int s = eidx[e];
    const int d = eidx[E + e];
    const float norm = dinv[s] * dinv[d];
    const float2 v = ((const float2*)(xw + s * DIM))[lane];
    float* o = out + d * DIM + lane * 2;
    atomicAdd(o,     v.x * norm);
    atomicAdd(o + 1, v.y * norm);
}

// --------------------------- classifier dot --------------------------------
// log_s[i] = (x_out[i] . W_cls + b_cls) / tau   (feed sinkhorn directly)
__global__ void cls_kernel(const float* __restrict__ out,
                           const float* __restrict__ Wc,
                           const float* __restrict__ bc,
                           float* __restrict__ skl) {
    const int i = blockIdx.x * blockDim.x + threadIdx.x;
    const float4* r = (const float4*)(out + i * DIM);
    const float4* w = (const float4*)Wc;
    float acc = 0.0f;
#pragma unroll
    for (int j = 0; j < DIM / 4; ++j) {
        float4 a = r[j], b = w[j];
        acc += a.x * b.x + a.y * b.y + a.z * b.z + a.w * b.w;
    }
    skl[i] = (acc + bc[0]) * SK_INV_TAU;
}

// --------------------------- sinkhorn (256x256) ----------------------------
// Single block, 256 threads. Row pass: thread t owns row t. Col pass:
// thread t owns column t. Matrix (256KB) stays in L2. n1==n2 -> no padding.
__global__ void sinkhorn_kernel(float* __restrict__ L) {
    const int t = threadIdx.x;
    for (int it = 0; it < SK_ITERS; ++it) {
        // row logsumexp
        float mx = -1e30f;
        for (int j = 0; j < SKN; ++j) mx = fmaxf(mx, L[t * SKN + j]);
        float s = 0.0f;
        for (int j = 0; j < SKN; ++j) s += expf(L[t * SKN + j] - mx);
        float lse = mx + logf(s);
        for (int j = 0; j < SKN; ++j) L[t * SKN + j] -= lse;
        __syncthreads();
        // column logsumexp
        mx = -1e30f;
        for (int i = 0; i < SKN; ++i) mx = fmaxf(mx, L[i * SKN + t]);
        s = 0.0f;
        for (int i = 0; i < SKN; ++i) s += expf(L[i * SKN + t] - mx);
        lse = mx + logf(s);
        for (int i = 0; i < SKN; ++i) L[i * SKN + t] -= lse;
        __syncthreads();
    }
}

// --------------------------- final broadcast add ---------------------------
__global__ void final_kernel(float* __restrict__ out,
                             const float* __restrict__ L) {
    const int gid = blockIdx.x * blockDim.x + threadIdx.x;  // N*DIM threads
    out[gid] += expf(L[gid >> 6]);
}

// ---------------------------------------------------------------------------
extern "C" void kernel_launch(void* const* d_in, const int* in_sizes, int n_in,
                              void* d_out, int out_size, void* d_ws, size_t ws_size,
                              hipStream_t stream) {
    const float* x    = (const float*)d_in[0];
    const int*   eidx = (const int*)d_in[1];
    // d_in[2], d_in[3]: n1, n2 (fixed at 256 in the reference)
    const float* Wg = (const float*)d_in[4];
    const float* bg = (const float*)d_in[5];
    const float* Ws = (const float*)d_in[6];
    const float* bs = (const float*)d_in[7];
    const float* Wc = (const float*)d_in[8];
    const float* bc = (const float*)d_in[9];
    float* out = (float*)d_out;

    const int E = in_sizes[1] / 2;   // 4194304

    float* ws   = (float*)d_ws;
    float* xw   = ws;                        // NNODES*DIM  (16 MB)
    float* degv = xw + (size_t)NNODES * DIM; // NNODES      (deg -> dinv in place)
    float* skl  = degv + NNODES;             // SKN*SKN log-matrix

    init_deg_kernel<<<NNODES / 256, 256, 0, stream>>>(degv);
    deg_count_kernel<<<(E + 255) / 256, 256, 0, stream>>>(eidx, degv, E);
    dinv_kernel<<<NNODES / 256, 256, 0, stream>>>(degv);

    // 4096 waves (one per 16-row strip), 8 waves per block -> exact grid,
    // EXEC is all-1s for every WMMA.
    gemm_wmma_kernel<<<(NNODES / 16) / 8, 256, 0, stream>>>(
        x, Wg, bg, Ws, bs, degv, xw, out);

    scatter_kernel<<<(E * 32) / 256, 256, 0, stream>>>(eidx, xw, degv, out, E);

    cls_kernel<<<NNODES / 256, 256, 0, stream>>>(out, Wc, bc, skl);
    sinkhorn_kernel<<<1, 256, 0, stream>>>(skl);
    final_kernel<<<(NNODES * 64) / 256, 256, 0, stream>>>(out, skl);
}